// BDHTinyModel_64939905516283
// MI455X (gfx1250) — compile-verified
//
#include <hip/hip_runtime.h>
#include <hip/hip_bf16.h>

// ---------------- model dims (compile-time) ----------------
constexpr int BB   = 4;
constexpr int TT   = 1024;
constexpr int DD   = 512;
constexpr int NHH  = 8;
constexpr int NN   = 1024;      // D*MULT/NH
constexpr int VOC  = 32000;
constexpr int NLAYER = 4;
constexpr float EPSF = 1e-5f;

typedef __attribute__((ext_vector_type(2))) float v2f;
typedef __attribute__((ext_vector_type(8))) float v8f;
typedef __attribute__((ext_vector_type(4))) unsigned int v4u;
typedef __attribute__((ext_vector_type(8))) int v8i_t;
typedef __attribute__((ext_vector_type(4))) int v4i_t;

constexpr int EPI_NONE = 0;
constexpr int EPI_RELU = 1;
constexpr int EPI_RELU_MUL = 2;

// K staged in 16-element chunks, double buffered.
// LDS row stride (dwords) for K-major tiles: 16 K + 2 pad -> 8B aligned,
// and 18*lane mod 64 distinct for 16 lanes -> conflict-free column reads.
constexpr int KCH = 16;
constexpr int LDSTRIDE = 18;

// ---------------- TDM 2D tile load: global -> LDS ----------------
// Builds D# group0/group1 per CDNA5 ISA 8.3/8.4 and issues tensor_load_to_lds.
// tile_w elements contiguous per row (dim0), tile_h rows (dim1),
// row_stride in elements. pad=true inserts 2 dwords after every 16 dwords.
__device__ __forceinline__ void tdm_load_2d(unsigned lds_off, const float* gptr,
                                            int tile_w, int tile_h,
                                            long row_stride, bool pad) {
    unsigned long long ga = (unsigned long long)(uintptr_t)gptr;
    v4u g0;
    g0.x = 1u;                                     // count=1, user descriptor
    g0.y = lds_off;                                // lds_addr (bytes)
    g0.z = (unsigned)ga;                           // global_addr[31:0]
    g0.w = (unsigned)((ga >> 32) & 0x01FFFFFFu)    // global_addr[56:32]
         | (2u << 30);                             // type=2 ("image")

    unsigned w0 = (2u << 16);                      // data_size=2 -> 4 bytes
    if (pad) w0 |= (1u << 20)                      // pad_enable
              |  (3u << 22)                        // pad_interval: 16 dwords
              |  (1u << 25);                       // pad_amount: 2 dwords
    unsigned long long st = (unsigned long long)row_stride;
    v8i_t g1;
    g1[0] = (int)w0;
    g1[1] = (int)(((unsigned)tile_w & 0xFFFFu) << 16);   // tensor_dim0[15:0]
    g1[2] = (int)(((unsigned)tile_h & 0xFFFFu) << 16);   // tensor_dim1[15:0]
    g1[3] = (int)(((unsigned)tile_w & 0xFFFFu) << 16);   // tile_dim0
    g1[4] = (int)((unsigned)tile_h & 0xFFFFu);           // tile_dim1
    g1[5] = (int)(unsigned)(st & 0xFFFFFFFFull);         // dim0_stride[31:0]
    g1[6] = (int)(unsigned)((st >> 32) & 0xFFFFull);     // dim0_stride[47:32]
    g1[7] = 0;                                           // dim1_stride (2D)

    v4i_t gz = {0, 0, 0, 0};
#if __clang_major__ >= 23
    v8i_t gz8 = {};
    __builtin_amdgcn_tensor_load_to_lds(g0, g1, gz, gz, gz8, 0);
#else
    __builtin_amdgcn_tensor_load_to_lds(g0, g1, gz, gz, 0);
#endif
}

// ---------------- block reduction helper (wave32) ----------------
__device__ __forceinline__ float block_sum(float v) {
    __shared__ float sb[8];
    #pragma unroll
    for (int o = 16; o > 0; o >>= 1) v += __shfl_down(v, o, 32);
    int lane = threadIdx.x & 31, w = threadIdx.x >> 5;
    __syncthreads();
    if (lane == 0) sb[w] = v;
    __syncthreads();
    float tot = 0.f;
    #pragma unroll
    for (int i = 0; i < 8; ++i) tot += sb[i];
    return tot;
}

// ---------------- embed gather + layernorm ----------------
__global__ __launch_bounds__(256)
void k_embed_ln(const int* __restrict__ idx, const float* __restrict__ embed,
                float* __restrict__ v) {
    int row = blockIdx.x;
    int tok = idx[row];
    const float* src = embed + (long)tok * DD;
    float x0 = src[threadIdx.x], x1 = src[threadIdx.x + 256];
    float mu = block_sum(x0 + x1) * (1.f / DD);
    float d0 = x0 - mu, d1 = x1 - mu;
    float var = block_sum(d0 * d0 + d1 * d1) * (1.f / DD);
    float r = rsqrtf(var + EPSF);
    float* dst = v + (long)row * DD;
    dst[threadIdx.x] = d0 * r;
    dst[threadIdx.x + 256] = d1 * r;
}

// ---------------- layernorm over last dim D=512 ----------------
__global__ __launch_bounds__(256)
void k_ln512(const float* __restrict__ in, float* __restrict__ out) {
    int row = blockIdx.x;
    const float* src = in + (long)row * DD;
    float x0 = src[threadIdx.x], x1 = src[threadIdx.x + 256];
    float mu = block_sum(x0 + x1) * (1.f / DD);
    float d0 = x0 - mu, d1 = x1 - mu;
    float var = block_sum(d0 * d0 + d1 * d1) * (1.f / DD);
    float r = rsqrtf(var + EPSF);
    float* dst = out + (long)row * DD;
    dst[threadIdx.x] = d0 * r;
    dst[threadIdx.x + 256] = d1 * r;
}

// ---------------- v = ln(v + ln(tmp)) ----------------
__global__ __launch_bounds__(256)
void k_res_ln(const float* __restrict__ tmp, float* __restrict__ v) {
    int row = blockIdx.x;
    const float* tp = tmp + (long)row * DD;
    float* vp = v + (long)row * DD;
    float t0 = tp[threadIdx.x], t1 = tp[threadIdx.x + 256];
    float mu = block_sum(t0 + t1) * (1.f / DD);
    float d0 = t0 - mu, d1 = t1 - mu;
    float var = block_sum(d0 * d0 + d1 * d1) * (1.f / DD);
    float r = rsqrtf(var + EPSF);
    float w0 = vp[threadIdx.x] + d0 * r;
    float w1 = vp[threadIdx.x + 256] + d1 * r;
    float mu2 = block_sum(w0 + w1) * (1.f / DD);
    float e0 = w0 - mu2, e1 = w1 - mu2;
    float var2 = block_sum(e0 * e0 + e1 * e1) * (1.f / DD);
    float r2 = rsqrtf(var2 + EPSF);
    vp[threadIdx.x] = e0 * r2;
    vp[threadIdx.x + 256] = e1 * r2;
}

// ---------------- rope (interleaved pairs on last dim N=1024) ----------------
__global__ __launch_bounds__(256)
void k_rope(const float* __restrict__ x, float* __restrict__ o, long npairs) {
    long i = (long)blockIdx.x * 256 + threadIdx.x;
    if (i >= npairs) return;
    int p = (int)(i % (NN / 2));
    long rem = i / (NN / 2);
    int t = (int)(rem % TT);
    float freq = __expf(-((float)(2 * p) / (float)NN) * 9.210340371976184f);
    float ang = (float)t * freq;
    float s, c;
    sincosf(ang, &s, &c);
    float x1 = x[2 * i], x2 = x[2 * i + 1];
    o[2 * i]     = x1 * c - x2 * s;
    o[2 * i + 1] = x1 * s + x2 * c;
}

// ---------------- yf[b,t,h*N+n] = y[b,h,t,n] ----------------
__global__ __launch_bounds__(256)
void k_yf(const float* __restrict__ y, float* __restrict__ yf, long total) {
    long i = (long)blockIdx.x * 256 + threadIdx.x;
    if (i >= total) return;
    int n = (int)(i % NN);
    long r = i / NN;
    int t = (int)(r % TT);
    long r2 = r / TT;
    int h = (int)(r2 % NHH);
    int b = (int)(r2 / NHH);
    yf[(((long)b * TT + t) * (NHH * NN)) + (long)h * NN + n] = y[i];
}

// ---------------- TDM double-buffered batched WMMA fp32 GEMM ----------------
// Block = 8 waves computes a 128x128 C tile; wave (wm,wn) in 4x2 owns a
// 32x64 register tile = 2x4 accumulators of 16x16. K staged in 16-chunks,
// double buffered: wave 0 DMAs the A tile, wave 1 DMAs the B tile for chunk
// i+1 while all 8 waves compute chunk i; each issuing wave waits only its
// own TENSORcnt (<=1 keeps the prefetch in flight).
template<bool BT, int EPI, bool TRIL, bool KCAP>
__global__ __launch_bounds__(256)
void k_gemm(const float* __restrict__ A, int lda, long sAb, long sAh,
            const float* __restrict__ Bm, int ldb, long sBb, long sBh,
            float* __restrict__ C, int ldc, long sCb, long sCh,
            const float* __restrict__ X, long sXb, long sXh,
            int Nb, int K, int nh) {
    __shared__ float sA[2][128 * LDSTRIDE];
    __shared__ float sB[2][128 * LDSTRIDE];

    int z = blockIdx.y;
    int b = z / nh, h = z % nh;
    const float* Ap = A + (long)b * sAb + (long)h * sAh;
    const float* Bp = Bm + (long)b * sBb + (long)h * sBh;
    float* Cp = C + (long)b * sCb + (long)h * sCh;

    int mt = (int)blockIdx.x / Nb, nt = (int)blockIdx.x % Nb;
    int m0 = mt * 128, n0 = nt * 128;

    int lane = threadIdx.x;              // 0..31
    int wv = threadIdx.y;                // 0..7
    int wm = wv >> 1, wn = wv & 1;       // wave tile coords (4x2)
    int lane15 = lane & 15;
    int half = lane >> 4;
    int kg = half * 2;                   // K sub-group (0 or 2)

    v8f acc[2][4] = {};

    bool skip = TRIL && (n0 > m0);       // block fully above strict-lower part
    if (!skip) {
        int kEnd = K;
        if (KCAP) { int cap = m0 + 128; kEnd = cap < K ? cap : K; }
        int nchunks = kEnd / KCH;
        unsigned offA0 = (unsigned)(uintptr_t)(&sA[0][0]);
        unsigned offA1 = (unsigned)(uintptr_t)(&sA[1][0]);
        unsigned offB0 = (unsigned)(uintptr_t)(&sB[0][0]);
        unsigned offB1 = (unsigned)(uintptr_t)(&sB[1][0]);

        // prologue: stage chunk 0
        if (wv == 0) {
            tdm_load_2d(offA0, Ap + (long)m0 * lda, KCH, 128, lda, true);
        } else if (wv == 1) {
            if (BT) tdm_load_2d(offB0, Bp + (long)n0 * ldb, KCH, 128, ldb, true);
            else    tdm_load_2d(offB0, Bp + n0, 128, KCH, ldb, false);
        }

        for (int ic = 0; ic < nchunks; ++ic) {
            int cur = ic & 1;
            __syncthreads();             // spare buffer fully consumed
            if (wv == 0) {               // prefetch next A tile, wait current
                if (ic + 1 < nchunks) {
                    int kc = (ic + 1) * KCH;
                    tdm_load_2d(cur ? offA0 : offA1,
                                Ap + (long)m0 * lda + kc, KCH, 128, lda, true);
                    __builtin_amdgcn_s_wait_tensorcnt(1);
                } else {
                    __builtin_amdgcn_s_wait_tensorcnt(0);
                }
            } else if (wv == 1) {        // prefetch next B tile, wait current
                if (ic + 1 < nchunks) {
                    int kc = (ic + 1) * KCH;
                    if (BT) tdm_load_2d(cur ? offB0 : offB1,
                                        Bp + (long)n0 * ldb + kc, KCH, 128, ldb, true);
                    else    tdm_load_2d(cur ? offB0 : offB1,
                                        Bp + (long)kc * ldb + n0, 128, KCH, ldb, false);
                    __builtin_amdgcn_s_wait_tensorcnt(1);
                } else {
                    __builtin_amdgcn_s_wait_tensorcnt(0);
                }
            }
            __syncthreads();             // current chunk visible to all waves

            const float* cA = sA[cur];
            const float* cB = sB[cur];
            #pragma unroll
            for (int k = 0; k < KCH; k += 4) {
                v2f av[2], bv[4];
                #pragma unroll
                for (int i = 0; i < 2; ++i)
                    av[i] = *(const v2f*)&cA[(wm * 32 + i * 16 + lane15) * LDSTRIDE + k + kg];
                #pragma unroll
                for (int j = 0; j < 4; ++j) {
                    if (BT) {
                        bv[j] = *(const v2f*)&cB[(wn * 64 + j * 16 + lane15) * LDSTRIDE + k + kg];
                    } else {
                        int n = wn * 64 + j * 16 + lane15;
                        bv[j].x = cB[(k + kg) * 128 + n];
                        bv[j].y = cB[(k + kg + 1) * 128 + n];
                    }
                }
                #pragma unroll
                for (int i = 0; i < 2; ++i)
                    #pragma unroll
                    for (int j = 0; j < 4; ++j)
                        acc[i][j] = __builtin_amdgcn_wmma_f32_16x16x4_f32(
                            false, av[i], false, bv[j], (short)0, acc[i][j], false, false);
            }
        }
    }

    // ---------------- epilogue + store ----------------
    const float* Xp = nullptr;
    if constexpr (EPI == EPI_RELU_MUL) Xp = X + (long)b * sXb + (long)h * sXh;
    #pragma unroll
    for (int i = 0; i < 2; ++i) {
        int rbase = m0 + wm * 32 + i * 16 + half * 8;
        #pragma unroll
        for (int j = 0; j < 4; ++j) {
            int col = n0 + wn * 64 + j * 16 + lane15;
            #pragma unroll
            for (int r = 0; r < 8; ++r) {
                int row = rbase + r;
                float val = acc[i][j][r];
                if (TRIL && col >= row) val = 0.f;
                if constexpr (EPI == EPI_RELU) val = val > 0.f ? val : 0.f;
                if constexpr (EPI == EPI_RELU_MUL) {
                    val = val > 0.f ? val : 0.f;
                    val *= Xp[(long)row * ldc + col];
                }
                Cp[(long)row * ldc + col] = val;
            }
        }
    }
}

// ---------------- host-side orchestration ----------------
extern "C" void kernel_launch(void* const* d_in, const int* in_sizes, int n_in,
                              void* d_out, int out_size, void* d_ws, size_t ws_size,
                              hipStream_t stream) {
    const int*   idx   = (const int*)d_in[0];
    const float* embed = (const float*)d_in[1];
    const float* dec_x = (const float*)d_in[2];   // (NH, D, N)
    const float* dec_y = (const float*)d_in[3];   // (NH, D, N)
    const float* enc   = (const float*)d_in[4];   // (NH*N, D)
    const float* lm    = (const float*)d_in[5];   // (D, VOCAB)
    float* out = (float*)d_out;

    char* ws = (char*)d_ws;
    float* v   = (float*)(ws);                    //   8 MB (B,T,D)
    float* x   = (float*)(ws + (8L   << 20));     // 128 MB (B,NH,T,N); later yf
    float* qr  = (float*)(ws + (136L << 20));     // 128 MB (B,NH,T,N); later y
    float* sc  = (float*)(ws + (264L << 20));     // 128 MB (B,NH,T,T)
    float* a   = (float*)(ws + (392L << 20));     //  64 MB (B,NH,T,D)
    float* tmp = (float*)(ws + (456L << 20));     //   8 MB (B,T,D)

    dim3 gblk(32, 8);                             // 8 waves per block

    k_embed_ln<<<BB * TT, 256, 0, stream>>>(idx, embed, v);

    const long sTN = (long)TT * NN, sTD = (long)TT * DD, sTT = (long)TT * TT;
    const long pairs = (long)BB * NHH * TT * (NN / 2);
    const long yel = (long)BB * NHH * TT * NN;

    for (int l = 0; l < NLAYER; ++l) {
        // x = relu(v @ dec_x[h])           M=T x N=N, K=D
        k_gemm<false, EPI_RELU, false, false><<<dim3(64, BB * NHH), gblk, 0, stream>>>(
            v, DD, sTD, 0, dec_x, NN, 0, (long)DD * NN,
            x, NN, (long)NHH * sTN, sTN, nullptr, 0, 0, 8, DD, NHH);

        // qr = rope(x)
        k_rope<<<(unsigned)((pairs + 255) / 256), 256, 0, stream>>>(x, qr, pairs);

        // scores = tril(qr @ qr^T, -1)     M=T x N=T, K=N
        k_gemm<true, EPI_NONE, true, false><<<dim3(64, BB * NHH), gblk, 0, stream>>>(
            qr, NN, (long)NHH * sTN, sTN, qr, NN, (long)NHH * sTN, sTN,
            sc, TT, (long)NHH * sTT, sTT, nullptr, 0, 0, 8, NN, NHH);

        // a = scores @ v                   M=T x N=D, K=T (capped per row blk)
        k_gemm<false, EPI_NONE, false, true><<<dim3(32, BB * NHH), gblk, 0, stream>>>(
            sc, TT, (long)NHH * sTT, sTT, v, DD, sTD, 0,
            a, DD, (long)NHH * sTD, sTD, nullptr, 0, 0, 4, TT, NHH);

        // a = ln(a)  (row-wise, in place)
        k_ln512<<<BB * NHH * TT, 256, 0, stream>>>(a, a);

        // y = relu(a_ln @ dec_y[h]) * x    (into qr buffer)
        k_gemm<false, EPI_RELU_MUL, false, false><<<dim3(64, BB * NHH), gblk, 0, stream>>>(
            a, DD, (long)NHH * sTD, sTD, dec_y, NN, 0, (long)DD * NN,
            qr, NN, (long)NHH * sTN, sTN, x, (long)NHH * sTN, sTN, 8, DD, NHH);

        // yf[b,t,h*N+n] = y[b,h,t,n]       (into x buffer, x is dead)
        k_yf<<<(unsigned)((yel + 255) / 256), 256, 0, stream>>>(qr, x, yel);

        // tmp = yf @ encoder               M=B*T x N=D, K=NH*N
        k_gemm<false, EPI_NONE, false, false><<<dim3(128, 1), gblk, 0, stream>>>(
            x, NHH * NN, 0, 0, enc, DD, 0, 0,
            tmp, DD, 0, 0, nullptr, 0, 0, 4, NHH * NN, 1);

        // v = ln(v + ln(tmp))
        k_res_ln<<<BB * TT, 256, 0, stream>>>(tmp, v);
    }

    // logits = v @ lm_head                 M=B*T x N=VOCAB, K=D
    k_gemm<false, EPI_NONE, false, false><<<dim3(32 * 250, 1), gblk, 0, stream>>>(
        v, DD, 0, 0, lm, VOC, 0, 0,
        out, VOC, 0, 0, nullptr, 0, 0, 250, DD, 1);

    (void)in_sizes; (void)n_in; (void)out_size; (void)ws_size;
}